// shift_4C_38104949850162
// MI455X (gfx1250) — compile-verified
//
#include <hip/hip_runtime.h>

// Shift kernel for x:(B=64, C=256, H=56, W=56) fp32, NCHW contiguous.
//   group0 (c in [0,51)):    y[h] = h==0    ? 0 : x[h-1]   (roll +1 along H)
//   group1 (c in [51,102)):  y[w] = w==W-1  ? 0 : x[w+1]   (roll -1 along W)
//   group2 (c in [102,153)): y[h] = h==H-1  ? 0 : x[h+1]   (roll -1 along H)
//   group3 (c in [153,204)): y[w] = w==0    ? 0 : x[w-1]   (roll +1 along W)
//   group4 (c in [204,256)): y = x
//
// Pure streaming op: 411 MB total traffic, 0 FLOPs -> HBM-bound (~17.6 us at
// 23.3 TB/s). Strategy: b128 accesses via clang ext_vector float4, non-temporal
// TH hints, wave32 coalescing along W. No WMMA/TDM: no reuse or matrix math.

typedef __attribute__((ext_vector_type(4))) float v4f;

namespace {
constexpr int kB = 64;
constexpr int kC = 256;
constexpr int kH = 56;
constexpr int kW = 56;
constexpr int kG = 51;          // kC / 5
constexpr int kW4 = kW / 4;     // 14 float4 per row
constexpr int kNVec = kB * kC * kH * kW4;   // 12,845,056 float4 elements
constexpr int kBlock = 256;                 // 8 wave32
constexpr int kGrid = (kNVec + kBlock - 1) / kBlock;  // 50,176 (exact)
}  // namespace

__global__ __launch_bounds__(kBlock)
void shift5_kernel(const float* __restrict__ x, float* __restrict__ y) {
    const int i4 = blockIdx.x * kBlock + threadIdx.x;   // float4 index
    if (i4 >= kNVec) return;

    // element index of this thread's float4 = i4*4; decompose for group logic
    const int w4  = i4 % kW4;            // which float4 within the row
    const int row = i4 / kW4;            // (b*C + c)*H + h
    const int h   = row % kH;
    const int c   = (row / kH) % kC;
    const int base = i4 * 4;             // flat element offset of w = 4*w4

    v4f v;
    if (c < kG) {
        // roll +1 along H: zero first row, else copy row h-1 (aligned b128)
        if (h == 0) {
            v = (v4f)(0.f);
        } else {
            v = __builtin_nontemporal_load(
                    reinterpret_cast<const v4f*>(x + base - kW));
        }
    } else if (c < 2 * kG) {
        // roll -1 along W: y[w] = x[w+1], zero at w==W-1 (last lane of last f4)
        const float* p = x + base + 1;
        v.x = __builtin_nontemporal_load(p + 0);
        v.y = __builtin_nontemporal_load(p + 1);
        v.z = __builtin_nontemporal_load(p + 2);
        v.w = (w4 == kW4 - 1) ? 0.f : __builtin_nontemporal_load(p + 3);
    } else if (c < 3 * kG) {
        // roll -1 along H: zero last row, else copy row h+1 (aligned b128)
        if (h == kH - 1) {
            v = (v4f)(0.f);
        } else {
            v = __builtin_nontemporal_load(
                    reinterpret_cast<const v4f*>(x + base + kW));
        }
    } else if (c < 4 * kG) {
        // roll +1 along W: y[w] = x[w-1], zero at w==0 (first lane of first f4)
        const float* p = x + base - 1;
        v.x = (w4 == 0) ? 0.f : __builtin_nontemporal_load(p + 0);
        v.y = __builtin_nontemporal_load(p + 1);
        v.z = __builtin_nontemporal_load(p + 2);
        v.w = __builtin_nontemporal_load(p + 3);
    } else {
        // pass-through group (aligned b128)
        v = __builtin_nontemporal_load(reinterpret_cast<const v4f*>(x + base));
    }

    __builtin_nontemporal_store(v, reinterpret_cast<v4f*>(y) + i4);
}

extern "C" void kernel_launch(void* const* d_in, const int* in_sizes, int n_in,
                              void* d_out, int out_size, void* d_ws, size_t ws_size,
                              hipStream_t stream) {
    (void)in_sizes; (void)n_in; (void)d_ws; (void)ws_size; (void)out_size;
    const float* x = reinterpret_cast<const float*>(d_in[0]);
    float* y = reinterpret_cast<float*>(d_out);
    shift5_kernel<<<kGrid, kBlock, 0, stream>>>(x, y);
}